// InnerProductDecoder_31696858644804
// MI455X (gfx1250) — compile-verified
//
#include <hip/hip_runtime.h>
#include <hip/hip_bf16.h>
#include <cstdint>

typedef __attribute__((ext_vector_type(2))) float v2f;
typedef __attribute__((ext_vector_type(8))) float v8f;

#define D_FEAT 128

// One wave32 handles 16 edges:
//   A = 16 src rows (16 x K), B = 16 dst rows as columns (K x 16),
//   D = A*B accumulated over K=128 in chunks of 4 with V_WMMA_F32_16X16X4_F32,
//   diagonal of D = per-edge dot products (exact fp32).
__global__ __launch_bounds__(256) void edge_inner_product_sigmoid_wmma(
    const float* __restrict__ z1, const float* __restrict__ z2,
    const long long* __restrict__ edge_index, float* __restrict__ out,
    long long n_edges)
{
    const int lane = (int)(threadIdx.x & 31);
    const long long wave  = ((long long)blockIdx.x * blockDim.x + threadIdx.x) >> 5;
    const long long ebase = wave * 16;
    if (ebase >= n_edges) return;           // uniform across the wave -> EXEC stays all-1s

    // A-matrix 16x4 f32 layout: lanes 0-15 -> M=lane,   K = 0,1
    //                           lanes 16-31 -> M=lane-16, K = 2,3
    // B-matrix 4x16 f32 mirrors it with M<->N, so src/dst loads are symmetric.
    const int m    = lane & 15;             // which edge in the group of 16
    const int koff = (lane >> 4) << 1;      // 0 (lanes 0-15) or 2 (lanes 16-31)

    const long long e  = ebase + m;
    const long long si = edge_index[e];                 // src node
    const long long di = edge_index[n_edges + e];       // dst node

    const float* ap = z1 + si * (long long)D_FEAT + koff;
    const float* bp = z2 + di * (long long)D_FEAT + koff;

    v8f c0 = {};
    v8f c1 = {};
#pragma unroll
    for (int kc = 0; kc < D_FEAT; kc += 8) {
        v2f a0 = *(const v2f*)(ap + kc);
        v2f b0 = *(const v2f*)(bp + kc);
        v2f a1 = *(const v2f*)(ap + kc + 4);
        v2f b1 = *(const v2f*)(bp + kc + 4);
        // 8 args: (neg_a, A, neg_b, B, c_mod, C, reuse_a, reuse_b)
        c0 = __builtin_amdgcn_wmma_f32_16x16x4_f32(false, a0, false, b0,
                                                   (short)0, c0, false, false);
        c1 = __builtin_amdgcn_wmma_f32_16x16x4_f32(false, a1, false, b1,
                                                   (short)0, c1, false, false);
    }
    v8f c = c0 + c1;   // elementwise: same C/D layout, valid accumulator merge

    // C/D 16x16 f32 layout: VGPR v, lanes 0-15 hold (M=v, N=lane);
    //                       lanes 16-31 hold (M=v+8, N=lane-16).
    // Diagonal (i,i): i<8  -> VGPR=i,   lane=i      (lanes 0-7)
    //                 i>=8 -> VGPR=i-8, lane=i+16   (lanes 24-31)
    const bool have = (lane < 8) || (lane >= 24);
    const int  vidx = (lane < 8) ? lane : (lane - 24);
    float d = c[0];
#pragma unroll
    for (int v = 1; v < 8; ++v) d = (vidx == v) ? c[v] : d;

    if (have) {
        const int i = (lane < 8) ? lane : (lane - 16);
        out[ebase + i] = 1.0f / (1.0f + __expf(-d));
    }
}

extern "C" void kernel_launch(void* const* d_in, const int* in_sizes, int n_in,
                              void* d_out, int out_size, void* d_ws, size_t ws_size,
                              hipStream_t stream) {
    const float*     z1  = (const float*)d_in[0];
    const float*     z2  = (const float*)d_in[1];
    const long long* ei  = (const long long*)d_in[2];   // int64 edge_index [2, E]
    float*           out = (float*)d_out;

    const long long E = (long long)out_size;            // number of edges
    const int  threads         = 256;                   // 8 waves / block
    const long long edges_per_block = (threads / 32) * 16;  // 128
    const int  grid = (int)((E + edges_per_block - 1) / edges_per_block);

    edge_inner_product_sigmoid_wmma<<<grid, threads, 0, stream>>>(z1, z2, ei, out, E);
}